// GraphAttentionLayer_26319559590761
// MI455X (gfx1250) — compile-verified
//
#include <hip/hip_runtime.h>

typedef __attribute__((ext_vector_type(2))) float v2f;
typedef __attribute__((ext_vector_type(8))) float v8f;

#define TT    12
#define LDSP  65           // padded LDS row stride (floats) -> conflict-free
#define GAT_ALPHA 0.2f

// One 16x16 f32 output tile: D += A(16xK slice) * B, K=64 in steps of 4 using
// V_WMMA_F32_16X16X4_F32. smA row-major [64][LDSP], smB row-major [64][LDSP].
__device__ __forceinline__ v8f wmma_tile_f32(const float* __restrict__ smA,
                                             const float* __restrict__ smB,
                                             int m0, int n0, int lane, v8f c) {
  const int half = lane >> 4;     // 0: lanes 0-15, 1: lanes 16-31
  const int l15  = lane & 15;
#pragma unroll
  for (int k0 = 0; k0 < 64; k0 += 4) {
    const int ka = k0 + (half << 1);          // K = 0/1 or 2/3 of this step
    v2f a, b;
    a.x = smA[(m0 + l15) * LDSP + ka];        // A 16x4: M=l15, K=ka
    a.y = smA[(m0 + l15) * LDSP + ka + 1];    //          K=ka+1
    b.x = smB[ka * LDSP + n0 + l15];          // B 4x16: K=ka,   N=l15
    b.y = smB[(ka + 1) * LDSP + n0 + l15];    //         K=ka+1
    c = __builtin_amdgcn_wmma_f32_16x16x4_f32(false, a, false, b,
                                              (short)0, c, false, false);
  }
  return c;
}

// Kernel A: per (n,t) block: h = inp @ W (WMMA), g0 = h.a[0:64], g1 = h.a[64:128]
__global__ void __launch_bounds__(256)
gat_h_kernel(const float* __restrict__ inp, const float* __restrict__ W,
             const float* __restrict__ avec,
             float* __restrict__ h_ws, float* __restrict__ g0_ws,
             float* __restrict__ g1_ws) {
  __shared__ float sX[64 * LDSP];
  __shared__ float sW[64 * LDSP];
  __shared__ float sH[64 * LDSP];
  __shared__ float sA[128];

  const int bi   = blockIdx.x;        // n*12 + t
  const int tid  = threadIdx.x;
  const int lane = tid & 31;
  const int wave = tid >> 5;

  const float* Xg = inp + (size_t)bi * 4096;
#pragma unroll
  for (int i = 0; i < 16; ++i) {
    const int idx = tid + i * 256;
    const int r = idx >> 6, cc = idx & 63;
    sX[r * LDSP + cc] = Xg[idx];
    sW[r * LDSP + cc] = W[idx];
  }
  if (tid < 128) sA[tid] = avec[tid];
  __syncthreads();

  // 4x4 grid of 16x16 tiles; 8 waves, 2 M-tiles each
  const int nt = wave & 3;
#pragma unroll
  for (int j = 0; j < 2; ++j) {
    const int m0 = ((wave >> 2) * 2 + j) * 16;
    const int n0 = nt * 16;
    v8f c = {};
    c = wmma_tile_f32(sX, sW, m0, n0, lane, c);
    const int crow = m0 + ((lane >> 4) << 3);
    const int ccol = n0 + (lane & 15);
#pragma unroll
    for (int i = 0; i < 8; ++i) sH[(crow + i) * LDSP + ccol] = c[i];
  }
  __syncthreads();

  float* Hg = h_ws + (size_t)bi * 4096;
#pragma unroll
  for (int i = 0; i < 16; ++i) {
    const int idx = tid + i * 256;
    Hg[idx] = sH[(idx >> 6) * LDSP + (idx & 63)];
  }

  if (tid < 64) {                       // per-node attention scalars
    float s0 = 0.f, s1 = 0.f;
#pragma unroll 8
    for (int k = 0; k < 64; ++k) {
      const float hv = sH[tid * LDSP + k];
      s0 += hv * sA[k];
      s1 += hv * sA[64 + k];
    }
    g0_ws[bi * 64 + tid] = s0;
    g1_ws[bi * 64 + tid] = s1;
  }
}

// Kernel B: per (n,t) block: closed-form softmax rows -> attention,
// h' = attention @ h (WMMA), elu epilogue.
__global__ void __launch_bounds__(256)
gat_attn_kernel(const float* __restrict__ h_ws, const float* __restrict__ g0_ws,
                const float* __restrict__ g1_ws,
                float* __restrict__ out_h, float* __restrict__ out_att) {
  __shared__ float sH[64 * LDSP];
  __shared__ float sAtt[64 * LDSP];
  __shared__ float sg0[2][64];
  __shared__ float sg1[2][64];

  const int bi   = blockIdx.x;
  const int nn   = bi / TT;
  const int tt   = bi - nn * TT;
  const int tid  = threadIdx.x;
  const int lane = tid & 31;
  const int wave = tid >> 5;

  const float* Hg = h_ws + (size_t)bi * 4096;
#pragma unroll
  for (int i = 0; i < 16; ++i) {
    const int idx = tid + i * 256;
    sH[(idx >> 6) * LDSP + (idx & 63)] = Hg[idx];
  }
  if (tid < 128) {                          // scalars of the two coupled times
    const int b = tid >> 6, u = tid & 63;
    const int s = 2 * tt + b - (tt >= 6 ? 12 : 0);
    const int gi = (nn * TT + s) * 64 + u;
    sg0[b][u] = g0_ws[gi];
    sg1[b][u] = g1_ws[gi];
  }
  __syncthreads();

  if (tid < 64) {                           // one thread per attention row v
    const int v = tid, b = v >> 5, vm = v & 31;
    if (tt < 6) {                           // row has 2 distinct values
      float x0 = sg0[b][2 * vm] + sg1[b][2 * vm];
      float x1 = sg0[b][2 * vm + 1] + sg1[b][2 * vm + 1];
      x0 = x0 > 0.f ? x0 : GAT_ALPHA * x0;
      x1 = x1 > 0.f ? x1 : GAT_ALPHA * x1;
      const float m = fmaxf(x0, x1);
      const float e0 = expf(x0 - m), e1 = expf(x1 - m);
      const float inv = 1.0f / (32.0f * (e0 + e1));
      const float p0 = e0 * inv, p1 = e1 * inv;
#pragma unroll
      for (int w = 0; w < 32; ++w) {
        sAtt[v * LDSP + w]      = p0;
        sAtt[v * LDSP + 32 + w] = p1;
      }
    } else {                                // 32 distinct values, each twice
      float vals[32];
      float m = -3.402823466e38f;
#pragma unroll
      for (int wm = 0; wm < 32; ++wm) {
        float x = sg0[b][2 * wm] + sg1[b][2 * wm + 1];
        x = x > 0.f ? x : GAT_ALPHA * x;
        vals[wm] = x;
        m = fmaxf(m, x);
      }
      float sum = 0.f;
#pragma unroll
      for (int wm = 0; wm < 32; ++wm) {
        const float e = expf(vals[wm] - m);
        vals[wm] = e;
        sum += e;
      }
      const float inv = 1.0f / (2.0f * sum);
#pragma unroll
      for (int wm = 0; wm < 32; ++wm) {
        const float p = vals[wm] * inv;
        sAtt[v * LDSP + wm]      = p;
        sAtt[v * LDSP + 32 + wm] = p;
      }
    }
  }
  __syncthreads();

  float* Ag = out_att + (size_t)bi * 4096;
#pragma unroll
  for (int i = 0; i < 16; ++i) {
    const int idx = tid + i * 256;
    Ag[idx] = sAtt[(idx >> 6) * LDSP + (idx & 63)];
  }

  // h' = attention @ h with fused ELU
  float* Og = out_h + (size_t)bi * 4096;
  const int nt = wave & 3;
#pragma unroll
  for (int j = 0; j < 2; ++j) {
    const int m0 = ((wave >> 2) * 2 + j) * 16;
    const int n0 = nt * 16;
    v8f c = {};
    c = wmma_tile_f32(sAtt, sH, m0, n0, lane, c);
    const int crow = m0 + ((lane >> 4) << 3);
    const int ccol = n0 + (lane & 15);
#pragma unroll
    for (int i = 0; i < 8; ++i) {
      const float x = c[i];
      Og[(crow + i) * 64 + ccol] = x > 0.f ? x : (expf(x) - 1.0f);
    }
  }
}

extern "C" void kernel_launch(void* const* d_in, const int* in_sizes, int n_in,
                              void* d_out, int out_size, void* d_ws, size_t ws_size,
                              hipStream_t stream) {
  (void)in_sizes; (void)n_in; (void)out_size; (void)ws_size;
  const float* inp  = (const float*)d_in[0];
  // d_in[1] (adj) is dead code in the reference
  const float* W    = (const float*)d_in[2];
  const float* avec = (const float*)d_in[3];
  float* out   = (float*)d_out;                 // [0..786432): elu(h'), then attention
  float* h_ws  = (float*)d_ws;                  // 192*4096 floats
  float* g0_ws = h_ws + 192 * 4096;             // 192*64 floats
  float* g1_ws = g0_ws + 192 * 64;              // 192*64 floats

  gat_h_kernel<<<192, 256, 0, stream>>>(inp, W, avec, h_ws, g0_ws, g1_ws);
  gat_attn_kernel<<<192, 256, 0, stream>>>(h_ws, g0_ws, g1_ws, out, out + 786432);
}